// SelfAttention_86225763435243
// MI455X (gfx1250) — compile-verified
//
#include <hip/hip_runtime.h>
#include <cmath>

#define EDIM 300
#define SDIM 2048
#define BDIM 64

typedef __attribute__((ext_vector_type(2))) float v2f;
typedef __attribute__((ext_vector_type(8))) float v8f;

// ---------------------------------------------------------------------------
// Fused attention pooling pass (used for both stages):
//   scores[s] = scale * dot(vec_b, kv[b,s])   (kv gathered from embedding, L2-resident)
//   a = softmax(scores);  h_out[b,:] = sum_s a[s] * kv[b,s,:]
// One block per batch row; scores + token indices cached in LDS.
// ---------------------------------------------------------------------------
__global__ __launch_bounds__(256) void attn_pool_kernel(
    const int* __restrict__ x, const float* __restrict__ emb,
    const float* __restrict__ vec, int vstride, float scale,
    float* __restrict__ h_out)
{
    __shared__ float sc[SDIM];    // scores, then unnormalized probs
    __shared__ int   sidx[SDIM];  // gathered token ids
    __shared__ float lvec[EDIM];
    __shared__ float wredm[8];
    __shared__ float wreds[8];

    const int b    = blockIdx.x;
    const int tid  = threadIdx.x;
    const int lane = tid & 31;
    const int wave = tid >> 5;    // 8 waves of 32 (wave32)

    for (int e = tid; e < EDIM; e += 256) lvec[e] = vec[b * vstride + e];
    __syncthreads();

    // ---- Phase 1: scores. One wave per token; lanes split the E dimension.
    for (int t = wave; t < SDIM; t += 8) {
        const int idx = x[b * SDIM + t];
        const float* row = emb + (long)idx * EDIM;
        float p = 0.f;
        for (int e = lane; e < EDIM; e += 32) p += row[e] * lvec[e];
        for (int off = 16; off; off >>= 1) p += __shfl_xor(p, off, 32);
        if (lane == 0) { sc[t] = p * scale; sidx[t] = idx; }
    }
    __syncthreads();

    // ---- Phase 2: block softmax over SDIM scores.
    float m = -INFINITY;
    for (int t = tid; t < SDIM; t += 256) m = fmaxf(m, sc[t]);
    for (int off = 16; off; off >>= 1) m = fmaxf(m, __shfl_xor(m, off, 32));
    if (lane == 0) wredm[wave] = m;
    __syncthreads();
    m = wredm[0];
    for (int w = 1; w < 8; ++w) m = fmaxf(m, wredm[w]);

    float sum = 0.f;
    for (int t = tid; t < SDIM; t += 256) {
        float ev = __expf(sc[t] - m);
        sc[t] = ev;
        sum += ev;
    }
    for (int off = 16; off; off >>= 1) sum += __shfl_xor(sum, off, 32);
    if (lane == 0) wreds[wave] = sum;
    __syncthreads();
    sum = 0.f;
    for (int w = 0; w < 8; ++w) sum += wreds[w];
    const float inv = 1.0f / sum;

    // ---- Phase 3: weighted pooling. Threads own E slots; loop tokens.
    const int e0 = tid;
    const int e1 = tid + 256;
    float acc0 = 0.f, acc1 = 0.f;
    #pragma unroll 2
    for (int t = 0; t < SDIM; ++t) {
        const float* row = emb + (long)sidx[t] * EDIM;
        if (t + 4 < SDIM && e0 < EDIM)
            __builtin_prefetch(emb + (long)sidx[t + 4] * EDIM + e0, 0, 3);
        const float w_ = sc[t];
        if (e0 < EDIM) acc0 += w_ * row[e0];
        if (e1 < EDIM) acc1 += w_ * row[e1];
    }
    if (e0 < EDIM) h_out[b * EDIM + e0] = acc0 * inv;
    if (e1 < EDIM) h_out[b * EDIM + e1] = acc1 * inv;
}

// ---------------------------------------------------------------------------
// Generic fp32 WMMA GEMM with a branch-free K loop:
//   D(MxN) = [A0 | A1](MxK) * B(KxN) (+ bias)
// Requirements exploited (all satisfied by the call sites):
//   * M is a multiple of 16 -> no row guards.
//   * K is a multiple of 4; any "dead" K pad columns of A are ZERO, so B's
//     k-index can be safely clamped to [0, Kb-2] without affecting results.
//   * Out-of-range B columns only affect out-of-range D columns (never
//     stored), so nB is clamped to N-1 instead of predicated.
//   * K0 (split point between A0 and A1) is even, so an even-aligned k-pair
//     never straddles the split; the split is one base-pointer select.
// Stores: columns [N, Nstore) are written as 0 (zero-fills K-pad of the
// consumer); columns >= Nstore are skipped.
// One wave32 per 16x16 D tile; V_WMMA_F32_16X16X4_F32, K-loop unrolled x4.
// ---------------------------------------------------------------------------
template <int TRANSB>
__global__ __launch_bounds__(32) void wmma_gemm_f32(
    const float* __restrict__ A0, int lda0,
    const float* __restrict__ A1, int lda1, int K0,
    const float* __restrict__ Bm, int ldb, int Kb,
    const float* __restrict__ bias,
    float* __restrict__ D, int ldd,
    int M, int N, int Nstore, int K)
{
    const int n0   = blockIdx.x * 16;
    const int m0   = blockIdx.y * 16;
    const int lane = threadIdx.x;
    const int half = lane >> 4;
    const int lm   = lane & 15;
    const int mA   = m0 + lm;                       // always < M
    const int nB   = n0 + lm;
    const int nBc  = (nB < N) ? nB : (N - 1);       // clamped B column
    const int h2   = half * 2;
    const int KbM2 = Kb - 2;

    // Precomputed row bases: Ap[ka] is valid for both segments.
    const float* A0r = A0 + (size_t)mA * lda0;
    const float* A1r = A1 + (size_t)mA * lda1 - K0;
    const float* Brc = Bm + (size_t)nBc * (TRANSB ? ldb : 1);

    v8f c = {};
    #pragma unroll 4
    for (int k = 0; k < K; k += 4) {
        const int ka = k + h2;
        const float* Ap = (ka < K0) ? A0r : A1r;    // base select, no branch
        v2f a;
        a.x = Ap[ka];
        a.y = Ap[ka + 1];

        const int kb = (ka < KbM2) ? ka : KbM2;     // in-bounds clamp
        v2f bf;
        if (TRANSB) {                               // B[k,n] = Bm[n*ldb + k]
            bf.x = Brc[kb];
            bf.y = Brc[kb + 1];
        } else {                                    // B[k,n] = Bm[k*ldb + n]
            bf.x = Brc[(size_t)kb * ldb];
            bf.y = Brc[(size_t)(kb + 1) * ldb];
        }

        // (neg_a, A, neg_b, B, c_mod, C, reuse_a, reuse_b); EXEC all-ones.
        c = __builtin_amdgcn_wmma_f32_16x16x4_f32(false, a, false, bf,
                                                  (short)0, c, false, false);
    }

    const int nOut = n0 + lm;
    if (nOut < Nstore) {
        const bool live = (nOut < N);
        const float bv = (bias && live) ? bias[nOut] : 0.f;
        #pragma unroll
        for (int v = 0; v < 8; ++v) {
            const int mOut = m0 + v + half * 8;
            D[(size_t)mOut * ldd + nOut] = live ? (c[v] + bv) : 0.f;
        }
    }
}

// ---------------------------------------------------------------------------
// Pipeline:
//  1) attn_pool (stage 1, shared query, scale 1/sqrt(E))            -> h1 [64,300]
//  2) WMMA: q2 = h1 @ W_att        [64x150, K=300]; stored [64,152], pad zeroed
//  3) WMMA: qW = q2 @ W_att^T      [64x300, K=152 padded, Kb=150]
//       (algebraic elimination of the reference's 11.8-GFLOP k2 GEMM)
//  4) attn_pool (stage 2, per-row qW, scale 1.0)                    -> h2 [64,300]
//  5) WMMA: scores = [h1|h2] @ W_out + b   [64x34, K=600, split at K0=300]
// ---------------------------------------------------------------------------
extern "C" void kernel_launch(void* const* d_in, const int* in_sizes, int n_in,
                              void* d_out, int out_size, void* d_ws, size_t ws_size,
                              hipStream_t stream) {
    const int*   x     = (const int*)d_in[0];     // [64, 2048]
    const float* emb   = (const float*)d_in[1];   // [50000, 300]
    const float* query = (const float*)d_in[2];   // [1, 300]
    const float* Watt  = (const float*)d_in[3];   // [300, 150]
    const float* Wout  = (const float*)d_in[4];   // [600, 34]
    const float* bias  = (const float*)d_in[5];   // [34]
    float* out = (float*)d_out;                   // [64, 34]

    float* ws = (float*)d_ws;
    float* h1 = ws;                    // 64*300
    float* h2 = h1 + BDIM * EDIM;      // 64*300
    float* q2 = h2 + BDIM * EDIM;      // 64*152 (K-padded, pad cols zeroed)
    float* qW = q2 + BDIM * 152;       // 64*300

    const float scale1 = 1.0f / sqrtf((float)EDIM);

    // Stage 1 pooling (vstride=0: all rows share the learned query)
    attn_pool_kernel<<<BDIM, 256, 0, stream>>>(x, emb, query, 0, scale1, h1);

    // q2 = h1 @ W_att : M=64 N=150 K=300; store ld=152 with cols 150..151 = 0
    wmma_gemm_f32<0><<<dim3(10, 4), 32, 0, stream>>>(
        h1, EDIM, h1, EDIM, 10000, Watt, 150, 300, nullptr,
        q2, 152, BDIM, 150, 152, 300);

    // qW = q2 @ W_att^T : M=64 N=300, K padded 150->152 (A pad is zero)
    wmma_gemm_f32<1><<<dim3(19, 4), 32, 0, stream>>>(
        q2, 152, q2, 152, 10000, Watt, 150, 150, nullptr,
        qW, EDIM, BDIM, EDIM, EDIM, 152);

    // Stage 2 pooling (per-row qW, no extra scale per reference)
    attn_pool_kernel<<<BDIM, 256, 0, stream>>>(x, emb, qW, EDIM, 1.0f, h2);

    // scores = [h1 | h2] @ W_out + b : M=64 N=34 K=600, split at K0=300
    wmma_gemm_f32<0><<<dim3(3, 4), 32, 0, stream>>>(
        h1, EDIM, h2, EDIM, EDIM, Wout, 34, 600, bias,
        out, 34, BDIM, 34, 34, 600);
}